// VectorQuantizer_54812372632218
// MI455X (gfx1250) — compile-verified
//
#include <hip/hip_runtime.h>

typedef float v2f __attribute__((ext_vector_type(2)));
typedef float v4f __attribute__((ext_vector_type(4)));
typedef float v8f __attribute__((ext_vector_type(8)));

#define N_TOK (16 * 4096)      // 65536 tokens
#define DIM 64
#define KCODES 1024
#define WAVES_PER_BLOCK 8
#define TOK_PER_WAVE 32        // two 16-row WMMA tiles per wave
#define TOK_PER_BLOCK (WAVES_PER_BLOCK * TOK_PER_WAVE)  // 256
#define NBLOCKS (N_TOK / TOK_PER_BLOCK)                 // 256

// ---------------------------------------------------------------------------
// Kernel 1: codebook squared norms  enorm[k] = sum_d cb[k][d]^2
// ---------------------------------------------------------------------------
__global__ void vq_enorm_kernel(const float* __restrict__ cb,
                                float* __restrict__ enorm) {
    int k = blockIdx.x * blockDim.x + threadIdx.x;
    if (k < KCODES) {
        const v4f* row = (const v4f*)(cb + (size_t)k * DIM);
        float s = 0.0f;
#pragma unroll
        for (int i = 0; i < DIM / 4; ++i) {
            v4f v = row[i];
            s = __builtin_fmaf(v.x, v.x, s);
            s = __builtin_fmaf(v.y, v.y, s);
            s = __builtin_fmaf(v.z, v.z, s);
            s = __builtin_fmaf(v.w, v.w, s);
        }
        enorm[k] = s;
    }
}

// ---------------------------------------------------------------------------
// Kernel 2: fused WMMA distances + argmin + gather + loss partials.
// One wave = two 16-token tiles (32 tokens) vs all 1024 codes.
// Two independent accumulator chains share every B fragment.
// ---------------------------------------------------------------------------
__global__ __launch_bounds__(256)
void vq_main_kernel(const float* __restrict__ x,
                    const float* __restrict__ cb,
                    const float* __restrict__ enorm,
                    float* __restrict__ qout,
                    float* __restrict__ idxout,
                    float* __restrict__ blockloss) {
    const int lane = threadIdx.x & 31;
    const int wave = threadIdx.x >> 5;
    const int half = lane >> 4;   // 0: lanes 0-15, 1: lanes 16-31
    const int m16  = lane & 15;
    const int tile = blockIdx.x * WAVES_PER_BLOCK + wave;   // 0..2047
    const int row0 = tile * TOK_PER_WAVE;                   // first of 32 rows

    const v2f* x2 = (const v2f*)x;    // x as float2: row stride 32 pairs
    const v2f* c2 = (const v2f*)cb;   // codebook as float2

    // A fragments for both row tiles, all 16 K-steps (K=4 each).
    // ISA 16x4 f32 A layout: lanes 0-15 hold K0/K1, lanes 16-31 hold K2/K3.
    // => lane needs {x[m][4s+2h], x[m][4s+2h+1]} per step s.
    v2f a0[16], a1[16];
#pragma unroll
    for (int s = 0; s < 16; ++s) {
        a0[s] = x2[(size_t)(row0 + m16) * 32 + 2 * s + half];
        a1[s] = x2[(size_t)(row0 + 16 + m16) * 32 + 2 * s + half];
    }

    float bestv0[8], bestv1[8];
    int   besti0[8], besti1[8];
#pragma unroll
    for (int e = 0; e < 8; ++e) {
        bestv0[e] = 3.4e38f; besti0[e] = 0;
        bestv1[e] = 3.4e38f; besti1[e] = 0;
    }

    for (int n0 = 0; n0 < KCODES; n0 += 16) {
        const int code = n0 + m16;
        // B fragment (4x16): lane n holds {cb[n][4s+2h], cb[n][4s+2h+1]}.
        // Preload all 16 fragments so loads batch into one clause.
        const v2f* brow = c2 + (size_t)code * 32 + half;
        v2f b[16];
#pragma unroll
        for (int s = 0; s < 16; ++s) b[s] = brow[2 * s];

        v8f acc0 = {};
        v8f acc1 = {};
#pragma unroll
        for (int s = 0; s < 16; ++s) {
            acc0 = __builtin_amdgcn_wmma_f32_16x16x4_f32(
                false, a0[s], false, b[s], (short)0, acc0, false, false);
            acc1 = __builtin_amdgcn_wmma_f32_16x16x4_f32(
                false, a1[s], false, b[s], (short)0, acc1, false, false);
        }

        const float en = enorm[code];
        // acc element e at this lane: (M = e + 8*half, N = code). Branchless
        // running argmin keeps EXEC all-ones around the WMMA region.
#pragma unroll
        for (int e = 0; e < 8; ++e) {
            float d0 = __builtin_fmaf(-2.0f, acc0[e], en);
            bool t0 = d0 < bestv0[e];
            bestv0[e] = t0 ? d0 : bestv0[e];
            besti0[e] = t0 ? code : besti0[e];
            float d1 = __builtin_fmaf(-2.0f, acc1[e], en);
            bool t1 = d1 < bestv1[e];
            bestv1[e] = t1 ? d1 : bestv1[e];
            besti1[e] = t1 ? code : besti1[e];
        }
    }

    // Min-reduce across the 16 lanes of each half (xor masks stay in-half).
    // Tie-break on smaller index to match jnp.argmin (first occurrence).
#pragma unroll
    for (int e = 0; e < 8; ++e) {
#pragma unroll
        for (int off = 8; off >= 1; off >>= 1) {
            float ov0 = __shfl_xor(bestv0[e], off, 32);
            int   oi0 = __shfl_xor(besti0[e], off, 32);
            bool t0 = (ov0 < bestv0[e]) || (ov0 == bestv0[e] && oi0 < besti0[e]);
            bestv0[e] = t0 ? ov0 : bestv0[e];
            besti0[e] = t0 ? oi0 : besti0[e];
            float ov1 = __shfl_xor(bestv1[e], off, 32);
            int   oi1 = __shfl_xor(besti1[e], off, 32);
            bool t1 = (ov1 < bestv1[e]) || (ov1 == bestv1[e] && oi1 < besti1[e]);
            bestv1[e] = t1 ? ov1 : bestv1[e];
            besti1[e] = t1 ? oi1 : besti1[e];
        }
        if (m16 == e) {
            idxout[row0 + e + 8 * half]      = (float)besti0[e];
            idxout[row0 + 16 + e + 8 * half] = (float)besti1[e];
        }
    }

    // Gather winning codebook rows -> quantized output (STE output equals the
    // gathered codes numerically) + commitment-loss partial.
    float losspart = 0.0f;
    v2f* q2 = (v2f*)qout;
#pragma unroll
    for (int r = 0; r < 16; ++r) {
        int idx_r = __shfl(besti0[r & 7], 16 * (r >> 3), 32);  // uniform bcast
        v2f q  = c2[(size_t)idx_r * 32 + lane];
        v2f xv = x2[(size_t)(row0 + r) * 32 + lane];
        float dx = q.x - xv.x;
        float dy = q.y - xv.y;
        losspart = __builtin_fmaf(dx, dx, losspart);
        losspart = __builtin_fmaf(dy, dy, losspart);
        __builtin_nontemporal_store(q, &q2[(size_t)(row0 + r) * 32 + lane]);
    }
#pragma unroll
    for (int r = 0; r < 16; ++r) {
        int idx_r = __shfl(besti1[r & 7], 16 * (r >> 3), 32);
        v2f q  = c2[(size_t)idx_r * 32 + lane];
        v2f xv = x2[(size_t)(row0 + 16 + r) * 32 + lane];
        float dx = q.x - xv.x;
        float dy = q.y - xv.y;
        losspart = __builtin_fmaf(dx, dx, losspart);
        losspart = __builtin_fmaf(dy, dy, losspart);
        __builtin_nontemporal_store(q, &q2[(size_t)(row0 + 16 + r) * 32 + lane]);
    }

    // Wave-level loss reduction, then block-level via LDS (fixed order).
#pragma unroll
    for (int off = 16; off >= 1; off >>= 1)
        losspart += __shfl_xor(losspart, off, 32);

    __shared__ float s_part[WAVES_PER_BLOCK];
    if (lane == 0) s_part[wave] = losspart;
    __syncthreads();
    if (threadIdx.x == 0) {
        float t = 0.0f;
#pragma unroll
        for (int w = 0; w < WAVES_PER_BLOCK; ++w) t += s_part[w];
        blockloss[blockIdx.x] = t;
    }
}

// ---------------------------------------------------------------------------
// Kernel 3: deterministic fixed-order reduction of 256 block partials.
// ---------------------------------------------------------------------------
__global__ void vq_loss_kernel(const float* __restrict__ blockloss,
                               float* __restrict__ out) {
    __shared__ float s[NBLOCKS];
    int t = threadIdx.x;
    s[t] = blockloss[t];
    __syncthreads();
    for (int step = NBLOCKS / 2; step >= 1; step >>= 1) {
        if (t < step) s[t] += s[t + step];
        __syncthreads();
    }
    if (t == 0)
        out[0] = 0.25f * s[0] / (float)((size_t)N_TOK * DIM);
}

// ---------------------------------------------------------------------------
extern "C" void kernel_launch(void* const* d_in, const int* in_sizes, int n_in,
                              void* d_out, int out_size, void* d_ws, size_t ws_size,
                              hipStream_t stream) {
    const float* x  = (const float*)d_in[0];   // [16,4096,64] f32
    const float* cb = (const float*)d_in[1];   // [1024,64]    f32

    float* out     = (float*)d_out;
    float* qout    = out;                              // N_TOK*64 floats
    float* idxout  = out + (size_t)N_TOK * DIM;        // N_TOK floats
    float* lossout = idxout + N_TOK;                   // 1 float

    float* enorm     = (float*)d_ws;                   // KCODES floats
    float* blockloss = enorm + KCODES;                 // NBLOCKS floats

    vq_enorm_kernel<<<(KCODES + 127) / 128, 128, 0, stream>>>(cb, enorm);
    vq_main_kernel<<<NBLOCKS, 256, 0, stream>>>(x, cb, enorm, qout, idxout,
                                                blockloss);
    vq_loss_kernel<<<1, NBLOCKS, 0, stream>>>(blockloss, lossout);
}